// ZGate_6992206758257
// MI455X (gfx1250) — compile-verified
//
#include <hip/hip_runtime.h>
#include <cstdint>

// ZGate diagonal complex phase: out_r = dr*xr - di*xi ; out_i = dr*xi + di*xr
// D = 3^12 = 531441 rows, B = 128 cols, fp32. Pure HBM-bound (~0.37 FLOP/B),
// floor ~47us at 23.3 TB/s. Strategy: 128-bit async global->LDS staging
// (gfx1250 async engine, ASYNCcnt) double-buffered per wave, scalar diag
// loads (wave-uniform row), non-temporal 128-bit stores.

typedef float v4f __attribute__((ext_vector_type(4)));  // native clang vector

#define BLOCK 256
#define GRID  4096
#define WAVES_PER_BLOCK (BLOCK / 32)

// Issue one per-lane 16B async copy global->LDS (GVS addressing:
// saddr = 64-bit base SGPR pair, vaddr = 32-bit unsigned byte offset,
// first operand = LDS byte offset VGPR).
#define ASYNC_B128(ldsoff, base, byteoff)                                  \
    asm volatile("global_load_async_to_lds_b128 %0, %1, %2"                \
                 :: "v"(ldsoff), "v"(byteoff), "s"(base) : "memory")

__global__ __launch_bounds__(BLOCK) void zgate_async_kernel(
    const float* __restrict__ xr,
    const float* __restrict__ xi,
    const float* __restrict__ dreal,
    const float* __restrict__ dimag,
    float* __restrict__ out,
    long long nvec)  // float4 count per plane = D*B/4 (multiple of 32)
{
    // smem[wave][buf][arr][lane], arr: 0=real 1=imag
    __shared__ v4f smem[WAVES_PER_BLOCK * 2 * 2 * 32];

    const int lane = threadIdx.x & 31;
    const int wave = threadIdx.x >> 5;

    const long long tid    = (long long)blockIdx.x * BLOCK + threadIdx.x;
    const long long stride = (long long)GRID * BLOCK;

    // LDS byte offsets for this lane's slots (low 32 bits of generic
    // pointer to addrspace(3) object == workgroup-relative LDS offset).
    uint32_t ldsR[2], ldsI[2];
    int      idxR[2], idxI[2];
#pragma unroll
    for (int b = 0; b < 2; ++b) {
        idxR[b] = ((wave * 2 + b) * 2 + 0) * 32 + lane;
        idxI[b] = ((wave * 2 + b) * 2 + 1) * 32 + lane;
        ldsR[b] = (uint32_t)(uintptr_t)&smem[idxR[b]];
        ldsI[b] = (uint32_t)(uintptr_t)&smem[idxI[b]];
    }

    if (tid >= nvec) return;  // whole wave uniform: nvec % 32 == 0

    v4f* __restrict__ outR = (v4f*)out;
    v4f* __restrict__ outI = (v4f*)(out + (size_t)nvec * 4);

    // Prologue: stage iteration 0 into buffer 0.
    {
        uint32_t off0 = (uint32_t)(tid * 16);
        ASYNC_B128(ldsR[0], xr, off0);
        ASYNC_B128(ldsI[0], xi, off0);
    }

    int buf = 0;
    for (long long v = tid; v < nvec; v += stride) {
        const long long vn   = v + stride;
        const bool      more = (vn < nvec);  // wave-uniform

        if (more) {
            // Buffer we are about to overwrite was consumed last iteration;
            // its ds_loads have retired (values were used), but make the
            // WAR against the async engine architecturally safe:
            asm volatile("s_wait_dscnt 0" ::: "memory");
            uint32_t offn = (uint32_t)(vn * 16);
            ASYNC_B128(ldsR[buf ^ 1], xr, offn);
            ASYNC_B128(ldsI[buf ^ 1], xi, offn);
            // 2 newly issued outstanding => current pair has landed.
            asm volatile("s_wait_asynccnt 2" ::: "memory");
        } else {
            asm volatile("s_wait_asynccnt 0" ::: "memory");
        }

        // Consume staged tile (ds_load_b128).
        v4f a  = smem[idxR[buf]];
        v4f bv = smem[idxI[buf]];

        // Row index is uniform across the wave (32 vec4 per row) ->
        // force scalar so diag comes in via s_load.
        int row = __builtin_amdgcn_readfirstlane((int)(v >> 5));
        float cr = dreal[row];
        float ci = dimag[row];

        v4f orv = cr * a - ci * bv;
        v4f oiv = cr * bv + ci * a;

        // Write-once 544MB stream: bypass cache retention with NT stores.
        __builtin_nontemporal_store(orv, outR + v);
        __builtin_nontemporal_store(oiv, outI + v);

        buf ^= 1;
    }
}

extern "C" void kernel_launch(void* const* d_in, const int* in_sizes, int n_in,
                              void* d_out, int out_size, void* d_ws, size_t ws_size,
                              hipStream_t stream) {
    const float* xr = (const float*)d_in[0];
    const float* xi = (const float*)d_in[1];
    const float* dr = (const float*)d_in[2];
    const float* di = (const float*)d_in[3];
    float* out = (float*)d_out;

    const long long nvec = (long long)in_sizes[0] / 4;  // D*B/4 float4 per plane

    zgate_async_kernel<<<GRID, BLOCK, 0, stream>>>(xr, xi, dr, di, out, nvec);
}